// WassOrderDistance_24163486007961
// MI455X (gfx1250) — compile-verified
//
#include <hip/hip_runtime.h>
#include <hip/hip_bf16.h>
#include <math.h>

typedef float v2f __attribute__((ext_vector_type(2)));
typedef float v8f __attribute__((ext_vector_type(8)));

#define NN 2048          // n == m
#define DD 1024          // feature dim
#define BM 128
#define BN 128
#define BK 32
#define LDA 36           // padded LDS stride (conflict-free 16x4 frag reads, 16B-aligned f4 stores)
#define INV_N (1.0 / 2048.0)
#define TOL 0.005
#define INV_SQRT_2PI 0.39894228040143267794
// k_ij underflows to exactly 0.0 (f64) for |i-j| >= 71 since exponent <= 500.01 - (i-j)^2/4 < -745.
// BAND=80 gives margin; all band-restricted loops below are exact, not approximate.
#define BAND 80

// ---------------- column stats: c[j] = mean_j / sqrt(var_j(ddof=1) + 1e-4) ----------------
__global__ __launch_bounds__(256) void colstats_kernel(const float* __restrict__ A,
                                                       float* __restrict__ c,
                                                       int rows, int cols) {
  int j = blockIdx.x * blockDim.x + threadIdx.x;
  if (j >= cols) return;
  double s1 = 0.0, s2 = 0.0;
  for (int i = 0; i < rows; ++i) {
    double x = (double)A[(size_t)i * cols + j];
    s1 += x; s2 += x * x;
  }
  double mean = s1 / rows;
  double var  = (s2 - s1 * s1 / rows) / (rows - 1);
  c[j] = (float)(mean / sqrt(var + 1e-4));
}

// ---------------- row squared norms of (A - c) : one wave per row ----------------
__global__ __launch_bounds__(256) void rownorm_kernel(const float* __restrict__ A,
                                                      const float* __restrict__ c,
                                                      float* __restrict__ sq,
                                                      int rows, int cols) {
  int r    = (int)((blockIdx.x * blockDim.x + threadIdx.x) >> 5);
  int lane = threadIdx.x & 31;
  if (r >= rows) return;
  float s = 0.0f;
  for (int k = lane; k < cols; k += 32) {
    float t = A[(size_t)r * cols + k] - c[k];
    s += t * t;
  }
  for (int off = 16; off; off >>= 1) s += __shfl_xor(s, off, 32);
  if (lane == 0) sq[r] = s;
}

// ---------------- WMMA GEMM + fused k/d epilogue ----------------
// d_ij = xsq_i + ysq_j - 2 * (Xn Yn^T)_ij          (f32, like the reference)
// k_ij = INV_SQRT_2PI * exp( -(i-j)^2/4 + (s_ij - d_ij)*10 )   (f64)
// Blocks entirely outside the diagonal band zero-fill k (exact: k underflows to 0 there)
// and skip the GEMM + exp entirely; d is never read off-band (only appears as k*d).
__global__ __launch_bounds__(256) void gemm_kd_kernel(const float* __restrict__ X,
                                                      const float* __restrict__ Y,
                                                      const float* __restrict__ cx,
                                                      const float* __restrict__ cy,
                                                      const float* __restrict__ xsq,
                                                      const float* __restrict__ ysq,
                                                      double* __restrict__ kmat,
                                                      float* __restrict__ dmat) {
  const int tid  = threadIdx.x;
  const int bm   = blockIdx.x * BM;
  const int bn   = blockIdx.y * BN;

  // Far-off-band block: min |i-j| over the 128x128 tile >= |bm-bn| - 127 >= BAND+1 > 70.
  int diff = bm - bn; if (diff < 0) diff = -diff;
  if (diff >= BM + BAND + 1) {
    // zero-fill the k tile: 128x128 doubles = 8192 double2, 256 threads x 32
    for (int t = tid; t < (BM * BN) / 2; t += 256) {
      int r = t >> 6;            // 0..127
      int c = (t & 63) * 2;      // 0..126
      double2 z; z.x = 0.0; z.y = 0.0;
      *(double2*)(kmat + (size_t)(bm + r) * NN + bn + c) = z;
    }
    return;
  }

  __shared__ float As[BM * LDA];
  __shared__ float Bs[BN * LDA];

  const int wid  = tid >> 5;
  const int lane = tid & 31;
  const int wm   = (wid & 1) * 64;   // wave row base in block tile (2 wave-rows)
  const int wn   = (wid >> 1) * 32;  // wave col base in block tile (4 wave-cols)
  const int lrow  = lane & 15;
  const int khalf = (lane >> 4) * 2; // K offset of this half-wave in 16x16x4 frag

  v8f acc[4][2] = {};                // 64x32 f32 output per wave

  for (int kt = 0; kt < DD; kt += BK) {
    // Stage A(=Xn) and B(=Yn) tiles: 128x32 each, 256 threads x 4 float4 phases
    #pragma unroll
    for (int ph = 0; ph < 4; ++ph) {
      int idx = tid + ph * 256;      // 0..1023
      int r   = idx >> 3;            // 0..127
      int c4  = (idx & 7) * 4;       // 0..28
      float4 xv = *(const float4*)(X + (size_t)(bm + r) * DD + kt + c4);
      float4 cv = *(const float4*)(cx + kt + c4);
      float4 av; av.x = xv.x - cv.x; av.y = xv.y - cv.y; av.z = xv.z - cv.z; av.w = xv.w - cv.w;
      *(float4*)(As + r * LDA + c4) = av;
      float4 yv = *(const float4*)(Y + (size_t)(bn + r) * DD + kt + c4);
      float4 dv = *(const float4*)(cy + kt + c4);
      float4 bv; bv.x = yv.x - dv.x; bv.y = yv.y - dv.y; bv.z = yv.z - dv.z; bv.w = yv.w - dv.w;
      *(float4*)(Bs + r * LDA + c4) = bv;
    }
    __syncthreads();

    #pragma unroll
    for (int kk = 0; kk < BK; kk += 4) {
      v2f af[4], bf[2];
      #pragma unroll
      for (int a = 0; a < 4; ++a)
        af[a] = *(const v2f*)(As + (wm + a * 16 + lrow) * LDA + kk + khalf);
      #pragma unroll
      for (int b = 0; b < 2; ++b)
        bf[b] = *(const v2f*)(Bs + (wn + b * 16 + lrow) * LDA + kk + khalf);
      #pragma unroll
      for (int a = 0; a < 4; ++a)
        #pragma unroll
        for (int b = 0; b < 2; ++b)
          acc[a][b] = __builtin_amdgcn_wmma_f32_16x16x4_f32(
              false, af[a], false, bf[b], (short)0, acc[a][b], false, false);
    }
    __syncthreads();
  }

  // Epilogue: C/D layout -> row = p + 8*(lane>>4), col = lane&15 per 16x16 tile
  const int rowhi = (lane >> 4) * 8;
  #pragma unroll
  for (int a = 0; a < 4; ++a) {
    #pragma unroll
    for (int b = 0; b < 2; ++b) {
      int gj = bn + wn + b * 16 + lrow;
      float ysqj = ysq[gj];
      #pragma unroll
      for (int p = 0; p < 8; ++p) {
        int gi = bm + wm + a * 16 + rowhi + p;
        float dv = xsq[gi] + ysqj - 2.0f * acc[a][b][p];
        int   dj = gi - gj;
        double dif = (double)dj * INV_N;
        double sv  = 50.0 / (dif * dif + 1.0);
        double ex  = -(double)(dj * dj) * 0.25 + (sv - (double)dv) * 10.0;
        size_t off = (size_t)gi * NN + gj;
        // exp(x) == 0.0 exactly for x <= -800 in f64; skip the transcendental there
        kmat[off] = (ex > -800.0) ? exp(ex) * INV_SQRT_2PI : 0.0;
        dmat[off] = dv;
      }
    }
  }
}

// ---------------- init: u = 1/n, flag = 0, dis = 0 ----------------
__global__ void init_kernel(double* __restrict__ u, int* __restrict__ flag,
                            double* __restrict__ dis) {
  int i = blockIdx.x * blockDim.x + threadIdx.x;
  if (i < NN) u[i] = INV_N;
  if (i == 0) { *flag = 0; *dis = 0.0; }
}

// ---------------- w = k^T u  (banded column reduction, coalesced over j) ----------------
__global__ __launch_bounds__(256) void colmv_kernel(const double* __restrict__ kmat,
                                                    const double* __restrict__ u,
                                                    double* __restrict__ w,
                                                    const int* __restrict__ flag, int gated) {
  if (gated && *flag) return;
  int j0 = blockIdx.x * 256;
  int j  = j0 + threadIdx.x;
  // rows outside [j0-BAND, j0+255+BAND] have k == 0 for every column of this block
  int ilo = (j0 > BAND) ? j0 - BAND : 0;
  int ihi = j0 + 256 + BAND; if (ihi > NN) ihi = NN;
  double s = 0.0;
  for (int i = ilo; i < ihi; ++i) s += kmat[(size_t)i * NN + j] * u[i];
  w[j] = s;
}

// ---------------- u = 1 / (n * k @ t),  t = v (use_v) or b/w ; banded row loop ----------------
__global__ __launch_bounds__(256) void rowmv_kernel(const double* __restrict__ kmat,
                                                    const double* __restrict__ w,
                                                    const double* __restrict__ v,
                                                    double* __restrict__ u,
                                                    const int* __restrict__ flag,
                                                    int gated, int use_v) {
  if (gated && *flag) return;
  __shared__ double t[NN];
  for (int j = threadIdx.x; j < NN; j += 256)
    t[j] = use_v ? v[j] : INV_N / w[j];
  __syncthreads();
  int r    = blockIdx.x * 8 + (threadIdx.x >> 5);
  int lane = threadIdx.x & 31;
  int lo = (r > BAND) ? r - BAND : 0;
  int hi = r + BAND + 1; if (hi > NN) hi = NN;
  double s = 0.0;
  for (int j = lo + lane; j < hi; j += 32) s += kmat[(size_t)r * NN + j] * t[j];
  for (int off = 16; off; off >>= 1) s += __shfl_xor(s, off, 32);
  if (lane == 0) u[r] = 1.0 / ((double)NN * s);
}

// ---------------- v = b / w ----------------
__global__ void vcompute_kernel(const double* __restrict__ w, double* __restrict__ v,
                                const int* __restrict__ flag, int gated) {
  if (gated && *flag) return;
  int j = blockIdx.x * blockDim.x + threadIdx.x;
  if (j < NN) v[j] = INV_N / w[j];
}

// ---------------- crit = sum |v*w - b| ; flag = (crit < tol) || isnan ----------------
__global__ __launch_bounds__(256) void crit_kernel(const double* __restrict__ v,
                                                   const double* __restrict__ w,
                                                   int* __restrict__ flag) {
  __shared__ double red[256];
  double s = 0.0;
  for (int j = threadIdx.x; j < NN; j += 256) s += fabs(v[j] * w[j] - INV_N);
  red[threadIdx.x] = s;
  __syncthreads();
  for (int st = 128; st; st >>= 1) {
    if (threadIdx.x < st) red[threadIdx.x] += red[threadIdx.x + st];
    __syncthreads();
  }
  if (threadIdx.x == 0) {
    double c = red[0];
    *flag = ((c < TOL) || isnan(c)) ? 1 : 0;
  }
}

// ---------------- dis += u_i * sum_j k_ij * d_ij * v_j  (wave per row, banded) ----------------
__global__ __launch_bounds__(256) void zrow_kernel(const double* __restrict__ kmat,
                                                   const float* __restrict__ dmat,
                                                   const double* __restrict__ v,
                                                   const double* __restrict__ u,
                                                   double* __restrict__ dis) {
  int r    = blockIdx.x * 8 + (threadIdx.x >> 5);
  int lane = threadIdx.x & 31;
  int lo = (r > BAND) ? r - BAND : 0;
  int hi = r + BAND + 1; if (hi > NN) hi = NN;
  double s = 0.0;
  for (int j = lo + lane; j < hi; j += 32) {
    size_t off = (size_t)r * NN + j;
    s += kmat[off] * (double)dmat[off] * v[j];
  }
  for (int off = 16; off; off >>= 1) s += __shfl_xor(s, off, 32);
  if (lane == 0) atomicAdd(dis, u[r] * s);
}

// ---------------- out[0] = dis ; out[1+idx] = u_i * k_ij * v_j (0 off-band) ----------------
__global__ __launch_bounds__(256) void tout_kernel(const double* __restrict__ kmat,
                                                   const double* __restrict__ u,
                                                   const double* __restrict__ v,
                                                   const double* __restrict__ dis,
                                                   float* __restrict__ out) {
  size_t idx = (size_t)blockIdx.x * blockDim.x + threadIdx.x;
  if (idx == 0) out[0] = (float)(*dis);
  if (idx < (size_t)NN * NN) {
    int i = (int)(idx >> 11);
    int j = (int)(idx & 2047);
    int dj = i - j; if (dj < 0) dj = -dj;
    out[1 + idx] = (dj <= BAND) ? (float)(u[i] * kmat[idx] * v[j]) : 0.0f;
  }
}

extern "C" void kernel_launch(void* const* d_in, const int* in_sizes, int n_in,
                              void* d_out, int out_size, void* d_ws, size_t ws_size,
                              hipStream_t stream) {
  (void)in_sizes; (void)n_in; (void)out_size; (void)ws_size;
  const float* X = (const float*)d_in[0];
  const float* Y = (const float*)d_in[1];
  float* out = (float*)d_out;

  const size_t KE = (size_t)NN * NN;
  char* base = (char*)d_ws;
  double* kmat = (double*)base;            base += KE * sizeof(double);   // 32 MB
  float*  dmat = (float*)base;             base += KE * sizeof(float);    // 16 MB
  double* u    = (double*)base;            base += NN * sizeof(double);
  double* v    = (double*)base;            base += NN * sizeof(double);
  double* w    = (double*)base;            base += NN * sizeof(double);
  double* dis  = (double*)base;            base += sizeof(double);
  int*    flag = (int*)base;               base += sizeof(double);        // keep 8B alignment
  float*  cx   = (float*)base;             base += DD * sizeof(float);
  float*  cy   = (float*)base;             base += DD * sizeof(float);
  float*  xsq  = (float*)base;             base += NN * sizeof(float);
  float*  ysq  = (float*)base;

  // stage 1: normalization constants and row norms
  colstats_kernel<<<DD / 256, 256, 0, stream>>>(X, cx, NN, DD);
  colstats_kernel<<<DD / 256, 256, 0, stream>>>(Y, cy, NN, DD);
  rownorm_kernel<<<NN / 8, 256, 0, stream>>>(X, cx, xsq, NN, DD);
  rownorm_kernel<<<NN / 8, 256, 0, stream>>>(Y, cy, ysq, NN, DD);

  // stage 2+3: WMMA GEMM with fused k/d epilogue (off-band blocks zero-fill only)
  gemm_kd_kernel<<<dim3(NN / BM, NN / BN), 256, 0, stream>>>(X, Y, cx, cy, xsq, ysq, kmat, dmat);

  // stage 4: Sinkhorn (pass 1 ungated, continuation gated by device flag)
  init_kernel<<<NN / 256, 256, 0, stream>>>(u, flag, dis);
  colmv_kernel<<<NN / 256, 256, 0, stream>>>(kmat, u, w, flag, 0);
  rowmv_kernel<<<NN / 8, 256, 0, stream>>>(kmat, w, v, u, flag, 0, 0);   // update 1
  colmv_kernel<<<NN / 256, 256, 0, stream>>>(kmat, u, w, flag, 0);
  vcompute_kernel<<<NN / 256, 256, 0, stream>>>(w, v, flag, 0);          // v = b/(k^T u)
  rowmv_kernel<<<NN / 8, 256, 0, stream>>>(kmat, w, v, u, flag, 0, 1);   // update 2
  colmv_kernel<<<NN / 256, 256, 0, stream>>>(kmat, u, w, flag, 0);
  crit_kernel<<<1, 256, 0, stream>>>(v, w, flag);

  for (int it = 0; it < 18; ++it) {                                      // updates 3..20
    colmv_kernel<<<NN / 256, 256, 0, stream>>>(kmat, u, w, flag, 1);
    rowmv_kernel<<<NN / 8, 256, 0, stream>>>(kmat, w, v, u, flag, 1, 0);
  }
  colmv_kernel<<<NN / 256, 256, 0, stream>>>(kmat, u, w, flag, 1);
  vcompute_kernel<<<NN / 256, 256, 0, stream>>>(w, v, flag, 1);
  rowmv_kernel<<<NN / 8, 256, 0, stream>>>(kmat, w, v, u, flag, 1, 1);   // update 21

  // stage 5: outputs
  zrow_kernel<<<NN / 8, 256, 0, stream>>>(kmat, dmat, v, u, dis);
  tout_kernel<<<(unsigned)(KE / 256), 256, 0, stream>>>(kmat, u, v, dis, out);
}